// KascadeAnchorAttention_53601191854757
// MI455X (gfx1250) — compile-verified
//
#include <hip/hip_runtime.h>
#include <hip/hip_bf16.h>
#include <math.h>
#include <stdint.h>

typedef float v2f __attribute__((ext_vector_type(2)));
typedef float v8f __attribute__((ext_vector_type(8)));
typedef unsigned int u32x4 __attribute__((ext_vector_type(4)));
typedef int i32x4 __attribute__((ext_vector_type(4)));
typedef int i32x8 __attribute__((ext_vector_type(8)));

#define B_   2
#define S_   2048
#define H_   16
#define HD_  64
#define D_   1024          // H_*HD_
#define SCALE_ 0.125f      // 1/sqrt(64)
#define NEG_BIG (-1.0e10f)

#if __has_builtin(__builtin_amdgcn_tensor_load_to_lds)
#define HAVE_TDM 1
#else
#define HAVE_TDM 0
#endif

__device__ __forceinline__ v8f wmma4(v2f a, v2f b, v8f c) {
  // D = A(16x4 f32) * B(4x16 f32) + C(16x16 f32)
  return __builtin_amdgcn_wmma_f32_16x16x4_f32(false, a, false, b, (short)0, c, false, false);
}

#if HAVE_TDM
// Issue a TDM DMA of a 2-D f32 tile (tile_w x tile_h) from global into LDS.
// D# layout per CDNA5 ISA 8.3/8.4: group0 {count, lds_addr, global_addr, type=2},
// group1 {data_size=4B, tensor dims, tile dims, dim0 stride}. Groups 2/3 zero (2-D).
__device__ __forceinline__ void tdm_load_2d_f32(
    const float* lds_ptr, const float* gptr,
    unsigned tensor_w, unsigned tensor_h, unsigned long long stride_elems,
    unsigned tile_w, unsigned tile_h) {
  const unsigned lds_off = (unsigned)(unsigned long long)(uintptr_t)lds_ptr;
  const unsigned long long ga = (unsigned long long)(uintptr_t)gptr;
  u32x4 g0;
  g0.x = 1u;                                                  // count=1 (valid, user)
  g0.y = lds_off;                                             // lds_addr (bytes)
  g0.z = (unsigned)(ga & 0xFFFFFFFFull);                      // global_addr[31:0]
  g0.w = (unsigned)((ga >> 32) & 0x01FFFFFFull) | (2u << 30); // ga[56:32] | type=2
  i32x8 g1;
  g1[0] = (int)(2u << 16);                                    // data_size = 4 bytes
  g1[1] = (int)((tensor_w & 0xFFFFu) << 16);                  // tensor_dim0 lo16 @bit48
  g1[2] = (int)(((tensor_w >> 16) & 0xFFFFu) | ((tensor_h & 0xFFFFu) << 16));
  g1[3] = (int)(((tensor_h >> 16) & 0xFFFFu) | ((tile_w & 0xFFFFu) << 16));
  g1[4] = (int)(tile_h & 0xFFFFu);                            // tile_dim1 (tile_dim2=0)
  g1[5] = (int)(stride_elems & 0xFFFFFFFFull);                // dim0_stride lo32
  g1[6] = (int)((stride_elems >> 32) & 0xFFFFull);            // dim0_stride hi16
  g1[7] = 0;
  i32x4 gz = {0, 0, 0, 0};
#if defined(__clang_major__) && (__clang_major__ >= 23)
  i32x8 gz8 = {0, 0, 0, 0, 0, 0, 0, 0};
  __builtin_amdgcn_tensor_load_to_lds(g0, g1, gz, gz, gz8, 0);
#else
  __builtin_amdgcn_tensor_load_to_lds(g0, g1, gz, gz, 0);
#endif
}
#endif  // HAVE_TDM

// ---------------------------------------------------------------------------
// Kernel 1: QKV projections. Block = 8 waves -> 128x64 output tile.
// Weight chunks (64x64) staged into LDS by the Tensor Data Mover, double
// buffered so DMA of chunk kc+1 overlaps WMMA compute on chunk kc.
// Each wave computes 16x64 (4 accumulators; A fragment reused 4x).
// Output scattered into [B, H, S, HD] layout.
// grid = 3 mats * (4096/128) * (1024/64) = 1536 blocks
// ---------------------------------------------------------------------------
__global__ __launch_bounds__(256) void qkv_gemm_kernel(
    const float* __restrict__ x,
    const float* __restrict__ wq, const float* __restrict__ wk,
    const float* __restrict__ wv,
    float* __restrict__ qws, float* __restrict__ kws, float* __restrict__ vws) {
  __shared__ float Bsh[2][64 * 64];
  const int tid  = threadIdx.x;
  const int lane = tid & 31;
  const int wvid = tid >> 5;                 // wave 0..7
  const int mat  = blockIdx.x / 512;
  const int rem  = blockIdx.x % 512;
  const int mb   = rem / 16, nb = rem % 16;  // M block (128 rows), N block (64 cols)
  const int half = lane >> 4, nl = lane & 15, koff = half * 2;

  const float* w    = (mat == 0) ? wq : ((mat == 1) ? wk : wv);
  float*       outp = (mat == 0) ? qws : ((mat == 1) ? kws : vws);
  const float* wtile = w + nb * 64;                       // top of this N stripe
  const float* pa    = x + (mb * 128 + wvid * 16 + nl) * D_ + koff;

#if HAVE_TDM
  if (wvid == 0)                                          // prologue: chunk 0
    tdm_load_2d_f32(&Bsh[0][0], wtile, D_, D_, D_, 64, 64);
#endif

  v8f acc0 = {}, acc1 = {}, acc2 = {}, acc3 = {};
  for (int kc = 0; kc < 16; ++kc) {
#if HAVE_TDM
    if (wvid == 0) {
      if (kc < 15) {
        tdm_load_2d_f32(&Bsh[(kc + 1) & 1][0], wtile + (kc + 1) * 64 * D_,
                        D_, D_, D_, 64, 64);
        __builtin_amdgcn_s_wait_tensorcnt(1);   // in-order TDM: chunk kc done
      } else {
        __builtin_amdgcn_s_wait_tensorcnt(0);
      }
    }
    __syncthreads();
    const float* Bl = &Bsh[kc & 1][0];
#else
    __syncthreads();                            // protect previous chunk's reads
    #pragma unroll
    for (int i = 0; i < 16; ++i) {
      const int idx = tid + i * 256;
      Bsh[0][idx] = wtile[(kc * 64 + (idx >> 6)) * D_ + (idx & 63)];
    }
    __syncthreads();
    const float* Bl = &Bsh[0][0];
#endif
    #pragma unroll
    for (int t = 0; t < 16; ++t) {
      v2f a = *reinterpret_cast<const v2f*>(pa + kc * 64 + 4 * t);
      const int r0 = 4 * t + koff;
      v2f b0, b1, b2, b3;
      b0.x = Bl[r0 * 64 +  0 + nl]; b0.y = Bl[(r0 + 1) * 64 +  0 + nl];
      b1.x = Bl[r0 * 64 + 16 + nl]; b1.y = Bl[(r0 + 1) * 64 + 16 + nl];
      b2.x = Bl[r0 * 64 + 32 + nl]; b2.y = Bl[(r0 + 1) * 64 + 32 + nl];
      b3.x = Bl[r0 * 64 + 48 + nl]; b3.y = Bl[(r0 + 1) * 64 + 48 + nl];
      acc0 = wmma4(a, b0, acc0);
      acc1 = wmma4(a, b1, acc1);
      acc2 = wmma4(a, b2, acc2);
      acc3 = wmma4(a, b3, acc3);
    }
#if HAVE_TDM
    __syncthreads();                            // buffer reuse guard
#endif
  }

  // N stripe is 64 cols aligned to one head: h == nb, d = nt*16+nl
  #pragma unroll
  for (int i = 0; i < 8; ++i) {
    const int gm = mb * 128 + wvid * 16 + i + 8 * half;   // row in [0,4096)
    const int b = gm >> 11, s = gm & 2047;
    float* o = outp + (((long)(b * H_ + nb) * S_ + s) * HD_);
    o[ 0 + nl] = acc0[i];
    o[16 + nl] = acc1[i];
    o[32 + nl] = acc2[i];
    o[48 + nl] = acc3[i];
  }
}

// ---------------------------------------------------------------------------
// Kernel 2: RoPE applied in-place to q and k ([B,H,S,HD]), one thread per pair
// ---------------------------------------------------------------------------
__global__ __launch_bounds__(256) void rope_kernel(
    float* __restrict__ qws, float* __restrict__ kws) {
  const int total = B_ * H_ * S_ * (HD_ / 2);
  int idx = blockIdx.x * blockDim.x + threadIdx.x;
  if (idx >= total) return;
  const int d2 = idx & 31;
  const int t  = idx >> 5;
  const int s  = t & (S_ - 1);
  const int bh = t >> 11;
  const float inv = __expf(-(float)(2 * d2) * (9.210340371976184f / 64.0f));
  const float ang = (float)s * inv;
  const float c = cosf(ang), sn = sinf(ang);
  const long base = ((long)(bh * S_ + s)) * HD_ + 2 * d2;
  {
    float e = qws[base], o = qws[base + 1];
    qws[base]     = e * c - o * sn;
    qws[base + 1] = e * sn + o * c;
  }
  {
    float e = kws[base], o = kws[base + 1];
    kws[base]     = e * c - o * sn;
    kws[base + 1] = e * sn + o * c;
  }
}

// ---------------------------------------------------------------------------
// Kernel 3: causal flash attention. One wave per (b,h,q-tile of 16 rows).
// 4 independent waves per block (no block-wide sync: trip counts differ).
// ---------------------------------------------------------------------------
__global__ __launch_bounds__(128) void flash_kernel(
    const float* __restrict__ qws, const float* __restrict__ kws,
    const float* __restrict__ vws, float* __restrict__ attn) {
  __shared__ float Pl[4][16 * 17];
  const int lane  = threadIdx.x & 31;
  const int wslot = threadIdx.x >> 5;
  const int wid   = blockIdx.x * 4 + wslot;
  const int qi    = wid & 127;
  const int bh    = wid >> 7;
  const int half  = lane >> 4, nl = lane & 15, koff = 2 * half;
  float* lp = &Pl[wslot][0];

  const float* qbase = qws + ((long)(bh * S_ + qi * 16 + nl)) * HD_ + koff;
  v2f qa[16];
  #pragma unroll
  for (int t = 0; t < 16; ++t) qa[t] = *reinterpret_cast<const v2f*>(qbase + 4 * t);

  v8f o0 = {}, o1 = {}, o2 = {}, o3 = {};
  float mrow[8], lrow[8];
  #pragma unroll
  for (int i = 0; i < 8; ++i) { mrow[i] = -3.0e38f; lrow[i] = 0.0f; }

  for (int j = 0; j <= qi; ++j) {
    const float* kb = kws + ((long)(bh * S_ + j * 16 + nl)) * HD_ + koff;
    v8f sacc = {};
    #pragma unroll
    for (int t = 0; t < 16; ++t) {
      v2f b = *reinterpret_cast<const v2f*>(kb + 4 * t);
      sacc = wmma4(qa[t], b, sacc);
    }

    #pragma unroll
    for (int i = 0; i < 8; ++i) {
      float sv = sacc[i] * SCALE_;
      if (j == qi && nl > i + 8 * half) sv = NEG_BIG;
      float mx = sv;
      mx = fmaxf(mx, __shfl_xor(mx, 1, 16));
      mx = fmaxf(mx, __shfl_xor(mx, 2, 16));
      mx = fmaxf(mx, __shfl_xor(mx, 4, 16));
      mx = fmaxf(mx, __shfl_xor(mx, 8, 16));
      const float mnew  = fmaxf(mrow[i], mx);
      const float p     = __expf(sv - mnew);
      float rs = p;
      rs += __shfl_xor(rs, 1, 16);
      rs += __shfl_xor(rs, 2, 16);
      rs += __shfl_xor(rs, 4, 16);
      rs += __shfl_xor(rs, 8, 16);
      const float alpha = __expf(mrow[i] - mnew);
      lrow[i] = lrow[i] * alpha + rs;
      mrow[i] = mnew;
      o0[i] *= alpha; o1[i] *= alpha; o2[i] *= alpha; o3[i] *= alpha;
      lp[(i + 8 * half) * 17 + nl] = p;
    }

    const float* vb = vws + ((long)(bh * S_ + j * 16)) * HD_;
    #pragma unroll
    for (int t = 0; t < 4; ++t) {
      v2f a; a.x = lp[nl * 17 + 4 * t + koff];
             a.y = lp[nl * 17 + 4 * t + koff + 1];
      const float* vr = vb + (4 * t + koff) * HD_;
      v2f b0, b1, b2, b3;
      b0.x = vr[ 0 + nl]; b0.y = vr[HD_ +  0 + nl];
      b1.x = vr[16 + nl]; b1.y = vr[HD_ + 16 + nl];
      b2.x = vr[32 + nl]; b2.y = vr[HD_ + 32 + nl];
      b3.x = vr[48 + nl]; b3.y = vr[HD_ + 48 + nl];
      o0 = wmma4(a, b0, o0);
      o1 = wmma4(a, b1, o1);
      o2 = wmma4(a, b2, o2);
      o3 = wmma4(a, b3, o3);
    }
  }

  const int b = bh >> 4, h = bh & 15;
  #pragma unroll
  for (int i = 0; i < 8; ++i) {
    const float inv = 1.0f / lrow[i];
    const int s = qi * 16 + i + 8 * half;
    float* op = attn + (((long)(b * S_ + s)) * H_ + h) * HD_;
    op[ 0 + nl] = o0[i] * inv;
    op[16 + nl] = o1[i] * inv;
    op[32 + nl] = o2[i] * inv;
    op[48 + nl] = o3[i] * inv;
  }
}

// ---------------------------------------------------------------------------
// Kernel 4: output projection attn[4096x1024] @ wo[1024x1024] -> d_out.
// Same TDM-staged structure as the QKV GEMM. grid = 32*16 = 512 blocks.
// ---------------------------------------------------------------------------
__global__ __launch_bounds__(256) void oproj_gemm_kernel(
    const float* __restrict__ attn, const float* __restrict__ wo,
    float* __restrict__ out) {
  __shared__ float Bsh[2][64 * 64];
  const int tid  = threadIdx.x;
  const int lane = tid & 31;
  const int wvid = tid >> 5;
  const int mb   = blockIdx.x / 16, nb = blockIdx.x % 16;
  const int half = lane >> 4, nl = lane & 15, koff = half * 2;

  const float* wtile = wo + nb * 64;
  const float* pa    = attn + (mb * 128 + wvid * 16 + nl) * D_ + koff;

#if HAVE_TDM
  if (wvid == 0)
    tdm_load_2d_f32(&Bsh[0][0], wtile, D_, D_, D_, 64, 64);
#endif

  v8f acc0 = {}, acc1 = {}, acc2 = {}, acc3 = {};
  for (int kc = 0; kc < 16; ++kc) {
#if HAVE_TDM
    if (wvid == 0) {
      if (kc < 15) {
        tdm_load_2d_f32(&Bsh[(kc + 1) & 1][0], wtile + (kc + 1) * 64 * D_,
                        D_, D_, D_, 64, 64);
        __builtin_amdgcn_s_wait_tensorcnt(1);
      } else {
        __builtin_amdgcn_s_wait_tensorcnt(0);
      }
    }
    __syncthreads();
    const float* Bl = &Bsh[kc & 1][0];
#else
    __syncthreads();
    #pragma unroll
    for (int i = 0; i < 16; ++i) {
      const int idx = tid + i * 256;
      Bsh[0][idx] = wtile[(kc * 64 + (idx >> 6)) * D_ + (idx & 63)];
    }
    __syncthreads();
    const float* Bl = &Bsh[0][0];
#endif
    #pragma unroll
    for (int t = 0; t < 16; ++t) {
      v2f a = *reinterpret_cast<const v2f*>(pa + kc * 64 + 4 * t);
      const int r0 = 4 * t + koff;
      v2f b0, b1, b2, b3;
      b0.x = Bl[r0 * 64 +  0 + nl]; b0.y = Bl[(r0 + 1) * 64 +  0 + nl];
      b1.x = Bl[r0 * 64 + 16 + nl]; b1.y = Bl[(r0 + 1) * 64 + 16 + nl];
      b2.x = Bl[r0 * 64 + 32 + nl]; b2.y = Bl[(r0 + 1) * 64 + 32 + nl];
      b3.x = Bl[r0 * 64 + 48 + nl]; b3.y = Bl[(r0 + 1) * 64 + 48 + nl];
      acc0 = wmma4(a, b0, acc0);
      acc1 = wmma4(a, b1, acc1);
      acc2 = wmma4(a, b2, acc2);
      acc3 = wmma4(a, b3, acc3);
    }
#if HAVE_TDM
    __syncthreads();
#endif
  }

  #pragma unroll
  for (int i = 0; i < 8; ++i) {
    float* o = out + (long)(mb * 128 + wvid * 16 + i + 8 * half) * D_ + nb * 64;
    o[ 0 + nl] = acc0[i];
    o[16 + nl] = acc1[i];
    o[32 + nl] = acc2[i];
    o[48 + nl] = acc3[i];
  }
}

// ---------------------------------------------------------------------------
// Kernel 5: last-token tile top-k. Softmax is monotonic -> rank raw logits.
// ---------------------------------------------------------------------------
__global__ __launch_bounds__(256) void topk_kernel(
    const float* __restrict__ qws, const float* __restrict__ kws,
    int* __restrict__ idx_out) {
  __shared__ float qv[64];
  __shared__ float logits[S_];
  __shared__ float scores[128];
  const int bh = blockIdx.x, tid = threadIdx.x;
  const float* q = qws + ((long)(bh * S_ + (S_ - 1))) * HD_;
  if (tid < 64) qv[tid] = q[tid];
  __syncthreads();
  for (int jj = tid * 8; jj < tid * 8 + 8; ++jj) {
    const float* kr = kws + ((long)(bh * S_ + jj)) * HD_;
    float acc = 0.0f;
    #pragma unroll
    for (int d = 0; d < 64; ++d) acc += qv[d] * kr[d];
    logits[jj] = acc * SCALE_;
  }
  __syncthreads();
  if (tid < 128) {
    float m = logits[tid * 16];
    #pragma unroll
    for (int u = 1; u < 16; ++u) m = fmaxf(m, logits[tid * 16 + u]);
    scores[tid] = m;
  }
  __syncthreads();
  if (tid == 0) {
    for (int r = 0; r < 8; ++r) {
      float best = scores[0]; int bi = 0;
      for (int t = 1; t < 128; ++t)
        if (scores[t] > best) { best = scores[t]; bi = t; }
      idx_out[bh * 8 + r] = bi;
      scores[bi] = -3.0e38f;
    }
  }
}

// ---------------------------------------------------------------------------
extern "C" void kernel_launch(void* const* d_in, const int* in_sizes, int n_in,
                              void* d_out, int out_size, void* d_ws, size_t ws_size,
                              hipStream_t stream) {
  (void)in_sizes; (void)n_in; (void)out_size; (void)ws_size;
  const float* x  = (const float*)d_in[0];
  const float* wq = (const float*)d_in[1];
  const float* wk = (const float*)d_in[2];
  const float* wv = (const float*)d_in[3];
  const float* wo = (const float*)d_in[4];
  float* out = (float*)d_out;

  float* ws   = (float*)d_ws;
  float* qws  = ws;                 // [B,H,S,HD]  4M floats
  float* kws  = ws + 4194304;       // [B,H,S,HD]  4M floats
  float* vws  = ws + 8388608;       // [B,H,S,HD]  4M floats
  float* attn = ws + 12582912;      // [B,S,H*HD]  4M floats

  qkv_gemm_kernel<<<1536, 256, 0, stream>>>(x, wq, wk, wv, qws, kws, vws);
  rope_kernel<<<(B_ * H_ * S_ * (HD_ / 2) + 255) / 256, 256, 0, stream>>>(qws, kws);
  flash_kernel<<<1024, 128, 0, stream>>>(qws, kws, vws, attn);
  oproj_gemm_kernel<<<512, 256, 0, stream>>>(attn, wo, out);
  topk_kernel<<<B_ * H_, 256, 0, stream>>>(qws, kws, (int*)(out + 4194304));
}